// PointSpatialTransformer_80393197846642
// MI455X (gfx1250) — compile-verified
//
#include <hip/hip_runtime.h>

// Problem constants from the reference.
#define HDIM 512
#define WDIM 512
#define PLANE (HDIM * WDIM)   // elements per flow channel plane

typedef float v4f __attribute__((ext_vector_type(4)));
typedef float v2f __attribute__((ext_vector_type(2)));

// -----------------------------------------------------------------------------
// Kernel 1: precompute L[h][w] = (nl1, nl0) from flow (1,2,H,W).
//   nl0 = 2*((h + flow[0,h,w])/511 - 0.5) = (h + f0) * (2/511) - 1
//   nl1 = 2*((w + flow[1,h,w])/511 - 0.5) = (w + f1) * (2/511) - 1
// Each thread handles 2 horizontally-adjacent pixels: b64 loads, b128 store.
// 2 MB table lives in d_ws and stays L2-resident for kernel 2.
// -----------------------------------------------------------------------------
__global__ __launch_bounds__(256) void build_L_kernel(const float* __restrict__ flow,
                                                      float* __restrict__ L) {
    int t   = blockIdx.x * 256 + threadIdx.x;   // 0 .. PLANE/2 - 1
    int pix = t << 1;                           // first of 2 pixels (same row: W even)
    int h   = pix >> 9;                         // pix / 512
    int w   = pix & 511;

    v2f f0 = *(const v2f*)(flow + pix);          // channel 0 (row displacement)
    v2f f1 = *(const v2f*)(flow + PLANE + pix);  // channel 1 (col displacement)

    const float s = 2.0f / 511.0f;
    float nl0a = fmaf((float)h       + f0.x, s, -1.0f);
    float nl0b = fmaf((float)h       + f0.y, s, -1.0f);
    float nl1a = fmaf((float)w       + f1.x, s, -1.0f);
    float nl1b = fmaf((float)(w + 1) + f1.y, s, -1.0f);

    v4f v = { nl1a, nl0a, nl1b, nl0b };          // L[...,0]=nl1, L[...,1]=nl0
    *(v4f*)(L + (pix << 1)) = v;
}

// -----------------------------------------------------------------------------
// Kernel 2: per point (x,y), bilinear (or nearest) sample of L, then rescale.
// 2 points per thread: b128 NT load of points, 4x b64 RT gathers per point
// (shared base address + immediate offsets 0 / 8 / 4096 / 4104 bytes),
// b128 NT store of output. Points are in [0, 510): x1,y1 <= 510 < 512, so no
// clamping is needed on the bilinear path (matches the reference exactly).
// -----------------------------------------------------------------------------
__global__ __launch_bounds__(256) void sample_kernel(const float* __restrict__ point,
                                                     const float* __restrict__ L,
                                                     const int* __restrict__ intep_ptr,
                                                     float* __restrict__ out,
                                                     int npts) {
    const int intep = *intep_ptr;               // uniform scalar read (no host sync)

    int  t  = blockIdx.x * 256 + threadIdx.x;
    long i0 = (long)t * 2;                      // first point index for this thread
    if (i0 >= npts) return;

    // Keep the streaming side ahead of gather latency: global_prefetch_b8,
    // ~128 KB ahead in the point stream (speculative; bounds-guarded anyway).
    if (i0 * 2 + 32768 < (long)npts * 2) {
        __builtin_prefetch(point + i0 * 2 + 32768, 0, 1);
    }

    bool two = (i0 + 1) < npts;
    v4f p;
    if (two) {
        p = __builtin_nontemporal_load((const v4f*)point + t);   // th:NT stream
    } else {
        v2f q = *(const v2f*)(point + i0 * 2);
        p = (v4f){ q.x, q.y, q.x, q.y };
    }

    float r[4];
#pragma unroll
    for (int k = 0; k < 2; ++k) {
        float x = k ? p.z : p.x;
        float y = k ? p.w : p.y;
        float o0, o1;
        if (intep) {
            float xt = truncf(x), yt = truncf(y);
            float xf = x - xt,    yf = y - yt;
            int   x0 = (int)xt,   y0 = (int)yt;

            const float* base = L + (((x0 << 9) + y0) << 1);
            v2f v00 = *(const v2f*)(base);            // L[x0 , y0 ]
            v2f v01 = *(const v2f*)(base + 2);        // L[x0 , y1 ]
            v2f v10 = *(const v2f*)(base + 2 * WDIM); // L[x1 , y0 ]
            v2f v11 = *(const v2f*)(base + 2 * WDIM + 2);

            float w00 = xf * yf;
            float w10 = xf * (1.0f - yf);
            float w01 = (1.0f - xf) * yf;
            float w11 = (1.0f - xf) * (1.0f - yf);

            float t0 = w00 * v00.x + w10 * v10.x + w01 * v01.x + w11 * v11.x;
            float t1 = w00 * v00.y + w10 * v10.y + w01 * v01.y + w11 * v11.y;
            o0 = (t1 + 1.0f) * 256.0f;   // (tmp[:,1]+1)/2*512
            o1 = (t0 + 1.0f) * 256.0f;   // (tmp[:,0]+1)/2*512
        } else {
            int xi = min((int)rintf(x), HDIM - 1);   // RNE rounding == jnp.round
            int yi = min((int)rintf(y), WDIM - 1);
            v2f v  = *(const v2f*)(L + (((xi << 9) + yi) << 1));
            o0 = (v.y + 1.0f) * 256.0f;
            o1 = (v.x + 1.0f) * 256.0f;
        }
        r[2 * k]     = o0;
        r[2 * k + 1] = o1;
    }

    if (two) {
        v4f o = { r[0], r[1], r[2], r[3] };
        __builtin_nontemporal_store(o, (v4f*)out + t);           // th:NT stream out
    } else {
        *(v2f*)(out + i0 * 2) = (v2f){ r[0], r[1] };
    }
}

// -----------------------------------------------------------------------------
// Launch: inputs are (point[1,N,2] f32, flow[1,2,512,512] f32, intep i32).
// d_ws holds the 2 MB L table.
// -----------------------------------------------------------------------------
extern "C" void kernel_launch(void* const* d_in, const int* in_sizes, int n_in,
                              void* d_out, int out_size, void* d_ws, size_t ws_size,
                              hipStream_t stream) {
    const float* point = (const float*)d_in[0];
    const float* flow  = (const float*)d_in[1];
    const int*   intep = (const int*)d_in[2];
    float* L   = (float*)d_ws;     // needs 512*512*2*4 = 2 MB of scratch
    float* out = (float*)d_out;

    int npts = in_sizes[0] / 2;    // point has N*2 elements

    // Kernel 1: 262144 pixels, 2 per thread -> 512 blocks of 256.
    build_L_kernel<<<(PLANE / 2 + 255) / 256, 256, 0, stream>>>(flow, L);

    // Kernel 2: 2 points per thread.
    int nthreads = (npts + 1) / 2;
    sample_kernel<<<(nthreads + 255) / 256, 256, 0, stream>>>(point, L, intep, out, npts);
}